// YoloLayer_57690000720321
// MI455X (gfx1250) — compile-verified
//
#include <hip/hip_runtime.h>
#include <cstdint>

// ---------------------------------------------------------------------------
// YOLO head for MI455X (gfx1250). Pure streaming transform, 355 MB traffic,
// ~15.2 us HBM floor @ 23.3 TB/s. No contraction -> WMMA inapplicable.
// CDNA5 paths used:
//   - v_tanh_f32 (1-TRANS sigmoid) to keep the TRANS pipe under the HBM roof
//   - global_store_async_from_lds_b128 drain (ASYNCcnt) for perfectly
//     lane-contiguous 512B-per-instruction output writes, with the shared
//     24-bit INST_OFFSET advancing LDS+global addresses in lockstep
// ---------------------------------------------------------------------------

#define GDIM   152
#define GG     (GDIM * GDIM)                  // 23104
#define NA     3
#define NCH    10                             // 7 + NUM_CLASSES
#define NB     64
#define VEC    4                              // gx cells per thread (152%4==0)
#define TPB    256
#define UNITS_PER_A   (GG / VEC)              // 5776
#define TOTAL_UNITS   (NB * NA * UNITS_PER_A) // 1108992
#define NBLOCKS       (TOTAL_UNITS / TPB)     // 4332 (exact)
#define TILE_BYTES    (TPB * NCH * VEC * 4)   // 40960 B per block

#if defined(__has_builtin)
# if __has_builtin(__builtin_amdgcn_tanhf)
#  define DEV_TANHF(x) __builtin_amdgcn_tanhf(x)
# elif __has_builtin(__builtin_amdgcn_tanh_f32)
#  define DEV_TANHF(x) __builtin_amdgcn_tanh_f32(x)
# endif
#endif

__device__ __forceinline__ float fast_sigmoid(float v) {
#ifdef DEV_TANHF
    // sigmoid(x) = 0.5*tanh(x/2) + 0.5  -> single TRANS op on CDNA5
    return __builtin_fmaf(DEV_TANHF(v * 0.5f), 0.5f, 0.5f);
#else
    float e = __builtin_amdgcn_exp2f(v * -1.44269504088896340736f);
    return __builtin_amdgcn_rcpf(1.0f + e);
#endif
}
__device__ __forceinline__ float fast_exp(float v) {
    return __builtin_amdgcn_exp2f(v * 1.44269504088896340736f);
}

__global__ __launch_bounds__(TPB)
void yolo_head_kernel(const float* __restrict__ x,
                      const int*   __restrict__ img_size,
                      float*       __restrict__ out) {
    __shared__ float smem[TPB * NCH * VEC];   // 40960 B, linear output layout

    const int tid = threadIdx.x;
    const int u   = blockIdx.x * TPB + tid;   // vec4-unit id

    // t2 = b*3 + a  (combined batch*anchor plane index, 0..191)
    const int t2  = u / UNITS_PER_A;
    const int s4  = u - t2 * UNITS_PER_A;     // vec4 unit within plane
    const int a   = t2 - 3 * (t2 / 3);        // anchor id 0..2
    const int gy  = s4 / (GDIM / VEC);        // s4 / 38
    const int gx  = (s4 - gy * (GDIM / VEC)) * VEC;

    const float stride = (float)img_size[0] * (1.0f / (float)GDIM);  // 4.0
    // exp(pw) * (ANCHOR/stride) * stride == exp(pw) * ANCHOR  (stride cancels)
    const float aw = (a == 0) ? 11.0f : (a == 1) ? 24.0f : 42.0f;
    const float ah = (a == 0) ? 14.0f : (a == 1) ? 17.0f : 27.0f;
    const float cx = (float)gx * stride;
    const float cy = (float)gy * stride;

    // ---- coalesced input loads: 10 channels, each a contiguous b128 -------
    // channel base = (b*30 + a*10) = 10 * t2
    const float* inp = x + ((size_t)t2 * NCH * GG + (size_t)(s4 * VEC));
    float4 p[NCH];
#pragma unroll
    for (int c = 0; c < NCH; ++c)
        p[c] = *reinterpret_cast<const float4*>(inp + (size_t)c * GG);

    // ---- transform --------------------------------------------------------
    float o[VEC * NCH];
#pragma unroll
    for (int j = 0; j < VEC; ++j) {
        float* q = &o[j * NCH];
        q[0] = __builtin_fmaf(fast_sigmoid(((const float*)&p[0])[j]), stride,
                              cx + (float)j * stride);
        q[1] = __builtin_fmaf(fast_sigmoid(((const float*)&p[1])[j]), stride, cy);
        q[2] = fast_exp(((const float*)&p[2])[j]) * aw;
        q[3] = fast_exp(((const float*)&p[3])[j]) * ah;
        q[4] = ((const float*)&p[4])[j];
        q[5] = ((const float*)&p[5])[j];
        q[6] = fast_sigmoid(((const float*)&p[6])[j]);
        q[7] = fast_sigmoid(((const float*)&p[7])[j]);
        q[8] = fast_sigmoid(((const float*)&p[8])[j]);
        q[9] = fast_sigmoid(((const float*)&p[9])[j]);
    }

    // ---- stage in LDS, linear (LDS image == output tile image) ------------
    float4* lvec = reinterpret_cast<float4*>(&smem[tid * (NCH * VEC)]);
#pragma unroll
    for (int k = 0; k < NCH; ++k)
        lvec[k] = *reinterpret_cast<const float4*>(&o[k * 4]);

    __syncthreads();   // barrier + dscnt wait: LDS tile complete

    // ---- drain 40KB tile: async LDS->global b128, shared immediate offset -
    // chunk c = k*256 + tid ; laddr = ldsbase + c*16 ; gaddr = gbase + c*16.
    // Both advance by 4096 per k -> encode as INST_OFFSET (added to BOTH).
    const unsigned laddr0 = (unsigned)(uintptr_t)&smem[0] + (unsigned)(tid * 16);
    const uint64_t gaddr0 = (uint64_t)out
                          + (uint64_t)blockIdx.x * TILE_BYTES
                          + (uint64_t)(tid * 16);
#define ASYNC_ST(K)                                                          \
    asm volatile("global_store_async_from_lds_b128 %0, %1, off offset:%2"    \
                 :: "v"(gaddr0), "v"(laddr0), "i"((K) * (TPB * 16))          \
                 : "memory")
    ASYNC_ST(0); ASYNC_ST(1); ASYNC_ST(2); ASYNC_ST(3); ASYNC_ST(4);
    ASYNC_ST(5); ASYNC_ST(6); ASYNC_ST(7); ASYNC_ST(8); ASYNC_ST(9);
#undef ASYNC_ST

    asm volatile("s_wait_asynccnt 0" ::: "memory");
}

extern "C" void kernel_launch(void* const* d_in, const int* in_sizes, int n_in,
                              void* d_out, int out_size, void* d_ws, size_t ws_size,
                              hipStream_t stream) {
    (void)in_sizes; (void)n_in; (void)d_ws; (void)ws_size; (void)out_size;
    const float* x   = (const float*)d_in[0];
    const int*   img = (const int*)d_in[1];
    float*       out = (float*)d_out;
    yolo_head_kernel<<<dim3(NBLOCKS), dim3(TPB), 0, stream>>>(x, img, out);
}